// GalerkinDecoder_86285892977134
// MI455X (gfx1250) — compile-verified
//
#include <hip/hip_runtime.h>

typedef __attribute__((ext_vector_type(2))) float v2f;
typedef __attribute__((ext_vector_type(8))) float v8f;

#define B_   16
#define N_   8192
#define DIN_ 128
#define WID_ 64
#define K_   64

__device__ __forceinline__ v8f wmma_f32(v2f a, v2f b, v8f c) {
  // D = A(16x4) * B(4x16) + C, exact fp32 -> v_wmma_f32_16x16x4_f32
  return __builtin_amdgcn_wmma_f32_16x16x4_f32(false, a, false, b, (short)0, c,
                                               false, false);
}

__device__ __forceinline__ float silu(float v) {
  return v / (1.f + __expf(-v));
}

// coeff[b, k, i] = sum_x act[b, x, i] * wbases[x, k]
// One block = (b, i-tile); each of 8 waves reduces a 1024-x slab and computes
// ALL 4 k-tiles (register blocking: act B-operand reused 4x per load pair).
__global__ __launch_bounds__(256) void proj_kernel(const float* __restrict__ act,
                                                   const float* __restrict__ wbases,
                                                   float* __restrict__ coeff,
                                                   int di) {
  __shared__ float red[8 * 4 * 8 * 32];   // 32 KB
  const int itiles = di >> 4;
  const int b  = blockIdx.x / itiles;
  const int i0 = (blockIdx.x % itiles) << 4;
  const int wave = threadIdx.x >> 5;
  const int lane = threadIdx.x & 31;
  const int hf   = lane >> 4;
  const int m    = lane & 15;
  const size_t abase = (size_t)b * N_ * di;

  v8f c0 = {}, c1 = {}, c2 = {}, c3 = {};
  const int xbeg = wave * (N_ / 8);
  for (int x = xbeg; x < xbeg + N_ / 8; x += 4) {
    const size_t r0 = (size_t)(x + 2 * hf) * K_ + m;      // wbases row base
    const size_t a0 = abase + (size_t)(x + 2 * hf) * di + i0 + m;
    __builtin_prefetch(act + a0 + 32 * di, 0, 1);          // stream-ahead hint
    v2f bb;                                                // B = act (shared)
    bb.x = act[a0];
    bb.y = act[a0 + di];
    v2f a;                                                 // A = wbases^T tiles
    a.x = wbases[r0];      a.y = wbases[r0 + K_];      c0 = wmma_f32(a, bb, c0);
    a.x = wbases[r0 + 16]; a.y = wbases[r0 + K_ + 16]; c1 = wmma_f32(a, bb, c1);
    a.x = wbases[r0 + 32]; a.y = wbases[r0 + K_ + 32]; c2 = wmma_f32(a, bb, c2);
    a.x = wbases[r0 + 48]; a.y = wbases[r0 + K_ + 48]; c3 = wmma_f32(a, bb, c3);
  }
  // deterministic cross-wave reduction through LDS
#pragma unroll
  for (int r = 0; r < 8; ++r) {
    red[((wave * 4 + 0) * 8 + r) * 32 + lane] = c0[r];
    red[((wave * 4 + 1) * 8 + r) * 32 + lane] = c1[r];
    red[((wave * 4 + 2) * 8 + r) * 32 + lane] = c2[r];
    red[((wave * 4 + 3) * 8 + r) * 32 + lane] = c3[r];
  }
  __syncthreads();
  if (wave == 0) {
#pragma unroll
    for (int t = 0; t < 4; ++t) {
#pragma unroll
      for (int r = 0; r < 8; ++r) {
        float s = 0.f;
#pragma unroll
        for (int w = 0; w < 8; ++w) s += red[((w * 4 + t) * 8 + r) * 32 + lane];
        const int M = (t << 4) + r + 8 * hf;
        coeff[((size_t)b * K_ + M) * di + i0 + m] = s;
      }
    }
  }
}

// coeff2[b, k, o] = sum_i coeff[b, k, i] * w[i, o, k]   (tiny: 17 MFLOP)
__global__ __launch_bounds__(64) void mix_kernel(const float* __restrict__ coeff,
                                                 const float* __restrict__ w,
                                                 float* __restrict__ coeff2,
                                                 int di) {
  const int bk = blockIdx.x;        // b*K + k
  const int k  = bk & (K_ - 1);
  const int o  = threadIdx.x;
  const float* crow = coeff + (size_t)bk * di;
  float acc = 0.f;
  for (int i = 0; i < di; ++i)
    acc += crow[i] * w[((size_t)i * WID_ + o) * K_ + k];
  coeff2[(size_t)bk * WID_ + o] = acc;
}

// actout[b,x,o] = silu( sum_k bases[x,k]*coeff2[b,k,o]
//                     + sum_i act[b,x,i]*lw[o,i] + lb[o] )
// 4 waves per block; each wave owns ONE 16-row x-tile and computes all 4
// o-tiles, so the A-operand (bases/act, b64 loads) is reused 4x per load.
__global__ __launch_bounds__(128) void recon_kernel(const float* __restrict__ act,
                                                    const float* __restrict__ bases,
                                                    const float* __restrict__ coeff2,
                                                    const float* __restrict__ lw,
                                                    const float* __restrict__ lb,
                                                    float* __restrict__ actout,
                                                    int di) {
  const int tile = blockIdx.x * 4 + (threadIdx.x >> 5);
  const int b    = tile >> 9;            // / 512 x-tiles
  const int x0   = (tile & 511) << 4;
  const int lane = threadIdx.x & 31;
  const int hf   = lane >> 4;
  const int m    = lane & 15;
  const size_t arow = ((size_t)b * N_ + x0 + m) * di;
  const size_t brow = (size_t)(x0 + m) * K_;

  v8f c0 = {}, c1 = {}, c2 = {}, c3 = {};
  // spectral reconstruction: reduce over K modes
#pragma unroll 4
  for (int kk = 0; kk < K_; kk += 4) {
    v2f a = *(const v2f*)(bases + brow + kk + 2 * hf);     // b64, shared by 4 tiles
    const size_t c2r = ((size_t)b * K_ + kk + 2 * hf) * WID_ + m;
    v2f bb;
    bb.x = coeff2[c2r];      bb.y = coeff2[c2r + WID_];      c0 = wmma_f32(a, bb, c0);
    bb.x = coeff2[c2r + 16]; bb.y = coeff2[c2r + WID_ + 16]; c1 = wmma_f32(a, bb, c1);
    bb.x = coeff2[c2r + 32]; bb.y = coeff2[c2r + WID_ + 32]; c2 = wmma_f32(a, bb, c2);
    bb.x = coeff2[c2r + 48]; bb.y = coeff2[c2r + WID_ + 48]; c3 = wmma_f32(a, bb, c3);
  }
  // residual path: reduce over di input channels, B = lw^T rows (b64 loads)
#pragma unroll 4
  for (int ii = 0; ii < di; ii += 4) {
    v2f a = *(const v2f*)(act + arow + ii + 2 * hf);
    const float* lwp = lw + ii + 2 * hf;
    v2f bb;
    bb = *(const v2f*)(lwp + (size_t)(m)      * di); c0 = wmma_f32(a, bb, c0);
    bb = *(const v2f*)(lwp + (size_t)(m + 16) * di); c1 = wmma_f32(a, bb, c1);
    bb = *(const v2f*)(lwp + (size_t)(m + 32) * di); c2 = wmma_f32(a, bb, c2);
    bb = *(const v2f*)(lwp + (size_t)(m + 48) * di); c3 = wmma_f32(a, bb, c3);
  }
  float* orow = actout + ((size_t)b * N_ + x0) * WID_;
#pragma unroll
  for (int r = 0; r < 8; ++r) {
    const int M = r + 8 * hf;
    orow[(size_t)M * WID_ + m]      = silu(c0[r] + lb[m]);
    orow[(size_t)M * WID_ + m + 16] = silu(c1[r] + lb[m + 16]);
    orow[(size_t)M * WID_ + m + 32] = silu(c2[r] + lb[m + 32]);
    orow[(size_t)M * WID_ + m + 48] = silu(c3[r] + lb[m + 48]);
  }
}

// out[b,x] = f2b + sum_o silu( sum_i act[b,x,i]*f1w[o,i] + f1b[o] ) * f2w[o]
__global__ __launch_bounds__(128) void final_kernel(const float* __restrict__ act,
                                                    const float* __restrict__ f1w,
                                                    const float* __restrict__ f1b,
                                                    const float* __restrict__ f2w,
                                                    const float* __restrict__ f2b,
                                                    float* __restrict__ out) {
  __shared__ float h[4][16 * WID_];      // per-wave hidden tile, 16 KB
  const int wave = threadIdx.x >> 5;
  const int tile = blockIdx.x * 4 + wave;
  const int b    = tile >> 9;
  const int x0   = (tile & 511) << 4;
  const int lane = threadIdx.x & 31;
  const int hf   = lane >> 4;
  const int m    = lane & 15;
  const size_t arow = ((size_t)b * N_ + x0 + m) * WID_;

  v8f c0 = {}, c1 = {}, c2 = {}, c3 = {};
#pragma unroll 4
  for (int ii = 0; ii < WID_; ii += 4) {
    v2f a = *(const v2f*)(act + arow + ii + 2 * hf);
    const float* wp = f1w + ii + 2 * hf;
    v2f bb;
    bb = *(const v2f*)(wp + (size_t)(m)      * WID_); c0 = wmma_f32(a, bb, c0);
    bb = *(const v2f*)(wp + (size_t)(m + 16) * WID_); c1 = wmma_f32(a, bb, c1);
    bb = *(const v2f*)(wp + (size_t)(m + 32) * WID_); c2 = wmma_f32(a, bb, c2);
    bb = *(const v2f*)(wp + (size_t)(m + 48) * WID_); c3 = wmma_f32(a, bb, c3);
  }
#pragma unroll
  for (int r = 0; r < 8; ++r) {
    const int M = r + 8 * hf;
    h[wave][M * WID_ + m]      = silu(c0[r] + f1b[m]);
    h[wave][M * WID_ + m + 16] = silu(c1[r] + f1b[m + 16]);
    h[wave][M * WID_ + m + 32] = silu(c2[r] + f1b[m + 32]);
    h[wave][M * WID_ + m + 48] = silu(c3[r] + f1b[m + 48]);
  }
  __syncthreads();
  if (m < 16 && hf == 0) {               // lanes 0..15 of each wave
    float acc = f2b[0];
#pragma unroll
    for (int o = 0; o < WID_; ++o) acc += h[wave][m * WID_ + o] * f2w[o];
    out[(size_t)b * N_ + x0 + m] = acc;
  }
}

extern "C" void kernel_launch(void* const* d_in, const int* in_sizes, int n_in,
                              void* d_out, int out_size, void* d_ws, size_t ws_size,
                              hipStream_t stream) {
  const float* x      = (const float*)d_in[0];
  const float* bases  = (const float*)d_in[1];
  const float* wbases = (const float*)d_in[2];
  const float* w[4]  = {(const float*)d_in[3], (const float*)d_in[6],
                        (const float*)d_in[9], (const float*)d_in[12]};
  const float* lw[4] = {(const float*)d_in[4], (const float*)d_in[7],
                        (const float*)d_in[10], (const float*)d_in[13]};
  const float* lb[4] = {(const float*)d_in[5], (const float*)d_in[8],
                        (const float*)d_in[11], (const float*)d_in[14]};
  const float* f1w = (const float*)d_in[15];
  const float* f1b = (const float*)d_in[16];
  const float* f2w = (const float*)d_in[17];
  const float* f2b = (const float*)d_in[18];
  float* out = (float*)d_out;

  float* ws     = (float*)d_ws;
  float* actA   = ws;                                   // B*N*WID
  float* actB   = actA   + (size_t)B_ * N_ * WID_;      // B*N*WID
  float* coeff  = actB   + (size_t)B_ * N_ * WID_;      // B*K*DIN (max)
  float* coeff2 = coeff  + (size_t)B_ * K_ * DIN_;      // B*K*WID

  const float* cur = x;
  float* nxt = actA;
  int di = DIN_;
  for (int li = 0; li < 4; ++li) {
    const int itiles = di >> 4;
    proj_kernel<<<B_ * itiles, 256, 0, stream>>>(cur, wbases, coeff, di);
    mix_kernel<<<B_ * K_, 64, 0, stream>>>(coeff, w[li], coeff2, di);
    recon_kernel<<<B_ * (N_ / 16) / 4, 128, 0, stream>>>(cur, bases, coeff2,
                                                         lw[li], lb[li], nxt, di);
    cur = nxt;
    nxt = (nxt == actA) ? actB : actA;
    di = WID_;
  }
  final_kernel<<<B_ * (N_ / 16) / 4, 128, 0, stream>>>(cur, f1w, f1b, f2w, f2b, out);
}